// GlobalAttentionPooling_68856915689481
// MI455X (gfx1250) — compile-verified
//
#include <hip/hip_runtime.h>
#include <math.h>

// Problem constants from the reference (D fixed at 128).
#define DIMS 128

typedef __attribute__((ext_vector_type(2))) float v2f;
typedef __attribute__((ext_vector_type(8))) float v8f;

// ---------------------------------------------------------------------------
// Kernel 1: gate[i] = dot(feat[i,:], w) + b  via V_WMMA_F32_16X16X4_F32.
// One wave32 per 16-node tile. B matrix = w[k0..k0+3] broadcast across all 16
// columns, so D[m][nn] == gate[m] for every nn; accumulate over K = 128 in 32
// WMMA steps.
//
// f32 WMMA fragment layouts (ISA 7.12.2):
//   A (16x4):  lanes 0-15 -> M=lane,   VGPR0=K0, VGPR1=K1
//              lanes16-31 -> M=lane-16,VGPR0=K2, VGPR1=K3
//   B (4x16):  symmetric: lanes 0-15 hold rows K0/K1, lanes 16-31 rows K2/K3
//   C/D (16x16 f32): VGPR r -> M = r + 8*(lane>=16), N = lane&15
//
// Tail handling: out-of-range rows are CLAMPED to row n-1 (valid memory);
// the garbage only affects D rows m >= n, which are never stored. This keeps
// the K-loop completely branch-free (no per-lane EXEC predication).
// ---------------------------------------------------------------------------
__global__ void gap_gate_wmma_kernel(const float* __restrict__ feat,
                                     const float* __restrict__ w,
                                     const float* __restrict__ b,
                                     float* __restrict__ gate,
                                     int n)
{
    const int wave = (int)((blockIdx.x * blockDim.x + threadIdx.x) >> 5);
    const int lane = (int)(threadIdx.x & 31);
    const int numTiles = (n + 15) >> 4;
    if (wave >= numTiles) return;   // wave-uniform: EXEC all-ones inside

    const int rowBase = wave << 4;
    const int m16 = lane & 15;          // M index this lane feeds for A
    const int hi  = lane >> 4;          // 0 -> K pair {0,1}, 1 -> K pair {2,3}
    int row = rowBase + m16;
    if (row >= n) row = n - 1;          // clamp: keeps loads in-bounds
    const float* frow = feat + (size_t)row * DIMS + 2 * hi;
    const float* wrow = w + 2 * hi;

    v8f c = {};
    #pragma unroll 8
    for (int k0 = 0; k0 < DIMS; k0 += 4) {
        v2f a, bb;
        a.x  = frow[k0];  a.y  = frow[k0 + 1];
        bb.x = wrow[k0];  bb.y = wrow[k0 + 1];
        // 8 args: (neg_a, A, neg_b, B, c_mod, C, reuse_a, reuse_b)
        c = __builtin_amdgcn_wmma_f32_16x16x4_f32(false, a, false, bb,
                                                  (short)0, c, false, false);
    }

    // Every column of D holds the same gates; lanes with N==0 (lanes 0 and 16)
    // write 8 rows each: lane 0 -> M 0..7, lane 16 -> M 8..15.
    if (m16 == 0) {
        const float gb = b[0];
        const int mbase = rowBase + 8 * hi;
        if (mbase + 7 < n) {
            float4 lo = make_float4(c[0] + gb, c[1] + gb, c[2] + gb, c[3] + gb);
            float4 hi4 = make_float4(c[4] + gb, c[5] + gb, c[6] + gb, c[7] + gb);
            *(float4*)(gate + mbase)     = lo;
            *(float4*)(gate + mbase + 4) = hi4;
        } else {
            for (int r = 0; r < 8; ++r) {
                const int m = mbase + r;
                if (m < n) gate[m] = c[r] + gb;
            }
        }
    }
}

// ---------------------------------------------------------------------------
// Kernel 2: one 128-thread block per segment. segment_ids are sorted, so the
// block binary-searches its contiguous [start,end) node range, reduces max and
// sum(exp) over the precomputed gates, then thread t accumulates
// sum_i alpha_i * feat[i][t] with alphas staged through LDS in 128-node
// chunks. feat reads are fully coalesced (128 lanes x 4B = one 512B row).
// ---------------------------------------------------------------------------
__global__ void gap_segment_pool_kernel(const float* __restrict__ feat,
                                        const float* __restrict__ gate,
                                        const int* __restrict__ segids,
                                        float* __restrict__ out,
                                        int n)
{
    const int g = (int)blockIdx.x;
    const int t = (int)threadIdx.x;   // 0..127, one feature dim per thread

    __shared__ int   s_bounds[2];
    __shared__ float s_red[DIMS];
    __shared__ float s_alpha[DIMS];

    if (t < 2) {
        // lower_bound(segids, g + t)
        const int target = g + t;
        int lo = 0, hi = n;
        while (lo < hi) {
            const int mid = (lo + hi) >> 1;
            if (segids[mid] < target) lo = mid + 1; else hi = mid;
        }
        s_bounds[t] = lo;
    }
    __syncthreads();
    const int start = s_bounds[0];
    const int end   = s_bounds[1];
    const int cnt   = end - start;

    if (cnt <= 0) {                       // empty segment -> zeros
        out[(size_t)g * DIMS + t] = 0.0f;
        return;                           // uniform across block
    }

    // ---- block max of gate over the segment ----
    float m = -__builtin_huge_valf();
    for (int i = start + t; i < end; i += DIMS) m = fmaxf(m, gate[i]);
    s_red[t] = m;
    __syncthreads();
    #pragma unroll
    for (int s = DIMS / 2; s > 0; s >>= 1) {
        if (t < s) s_red[t] = fmaxf(s_red[t], s_red[t + s]);
        __syncthreads();
    }
    const float gmax = s_red[0];
    __syncthreads();

    // ---- block sum of exp(gate - gmax) ----
    float se = 0.0f;
    for (int i = start + t; i < end; i += DIMS) se += expf(gate[i] - gmax);
    s_red[t] = se;
    __syncthreads();
    #pragma unroll
    for (int s = DIMS / 2; s > 0; s >>= 1) {
        if (t < s) s_red[t] += s_red[t + s];
        __syncthreads();
    }
    const float inv_denom = 1.0f / s_red[0];
    __syncthreads();

    // ---- weighted accumulation: acc = sum_i alpha_i * feat[i][t] ----
    float acc0 = 0.0f, acc1 = 0.0f, acc2 = 0.0f, acc3 = 0.0f;
    for (int i0 = start; i0 < end; i0 += DIMS) {
        const int cn = min(DIMS, end - i0);
        if (t < cn) s_alpha[t] = expf(gate[i0 + t] - gmax) * inv_denom;
        __syncthreads();

        const float* fp = feat + (size_t)i0 * DIMS + t;
        // Speculative prefetch of the next chunk (global_prefetch_b8);
        // safe even past the end of the buffer.
        __builtin_prefetch(fp + (size_t)DIMS * DIMS, 0, 1);

        int j = 0;
        for (; j + 3 < cn; j += 4) {
            acc0 = fmaf(s_alpha[j    ], fp[(size_t)(j    ) * DIMS], acc0);
            acc1 = fmaf(s_alpha[j + 1], fp[(size_t)(j + 1) * DIMS], acc1);
            acc2 = fmaf(s_alpha[j + 2], fp[(size_t)(j + 2) * DIMS], acc2);
            acc3 = fmaf(s_alpha[j + 3], fp[(size_t)(j + 3) * DIMS], acc3);
        }
        for (; j < cn; ++j)
            acc0 = fmaf(s_alpha[j], fp[(size_t)j * DIMS], acc0);
        __syncthreads();
    }

    out[(size_t)g * DIMS + t] = (acc0 + acc1) + (acc2 + acc3);
}

// ---------------------------------------------------------------------------
// Inputs (setup_inputs order): feat[N*128] f32, gate_w[128] f32, gate_b[1] f32,
// segment_ids[N] int, num_segments scalar. Output: readout[G*128] f32.
// Workspace: gate[N] floats (2 MB).
// ---------------------------------------------------------------------------
extern "C" void kernel_launch(void* const* d_in, const int* in_sizes, int n_in,
                              void* d_out, int out_size, void* d_ws, size_t ws_size,
                              hipStream_t stream) {
    const float* feat = (const float*)d_in[0];
    const float* gw   = (const float*)d_in[1];
    const float* gb   = (const float*)d_in[2];
    const int*   seg  = (const int*)d_in[3];
    float*       out  = (float*)d_out;
    float*       gate = (float*)d_ws;

    const int n = in_sizes[3];          // number of nodes
    const int G = out_size / DIMS;      // number of segments

    // Kernel 1: one wave (32 threads) per 16-node tile, 8 waves per block.
    const int tiles   = (n + 15) / 16;
    const int blocks1 = (tiles + 7) / 8;
    gap_gate_wmma_kernel<<<blocks1, 256, 0, stream>>>(feat, gw, gb, gate, n);

    // Kernel 2: one block per segment, 128 threads (one per feature dim).
    gap_segment_pool_kernel<<<G, DIMS, 0, stream>>>(feat, gate, seg, out, n);
}